// PoseVocab_15710990369689
// MI455X (gfx1250) — compile-verified
//
#include <hip/hip_runtime.h>
#include <stdint.h>

// ---------------- problem constants (match reference) ----------------
#define JJ   24      // joints
#define PPV  64      // pose vocab entries (P)
#define LL   32      // grid H = W
#define CC   16      // channels per plane
#define OUTC (JJ * 3 * CC)          // 1152 floats per point
#define PLANE_F (LL * LL * CC)      // 16384 floats = 64 KB per plane
#define PTS_PER_BLOCK 1024
#define RED_BLOCKS 240

typedef float        v4f  __attribute__((ext_vector_type(4)));
typedef unsigned int v4u  __attribute__((ext_vector_type(4)));
typedef int          v4i_t __attribute__((ext_vector_type(4)));
typedef int          v8i_t __attribute__((ext_vector_type(8)));

// ---------------- TDM: DMA one contiguous 64KB plane into LDS ----------------
#if __has_builtin(__builtin_amdgcn_tensor_load_to_lds)
#define HAVE_TDM 1
__device__ __forceinline__ void tdm_copy_plane(const float* gsrc, unsigned lds_byte_off) {
    unsigned long long ga = (unsigned long long)(uintptr_t)gsrc;
    // D# group0: count=1, lds_addr, global_addr[56:0], type=2
    v4u g0;
    g0.x = 1u;                                   // count = 1 valid descriptor
    g0.y = lds_byte_off;                         // LDS byte address
    g0.z = (unsigned)(ga & 0xffffffffu);         // global_addr[31:0]
    g0.w = (unsigned)((ga >> 32) & 0x01ffffffu)  // global_addr[56:32]
         | (2u << 30);                           // type = 2 ("image")
    // D# group1: data_size=4B; tensor_dim0=tile_dim0=16384 elems; dims1..2 = 1 row
    v8i_t g1;
    g1[0] = 0x00020000;  // [17:16] data_size = 2 (4 bytes); wg_mask = 0
    g1[1] = 0x40000000;  // tensor_dim0[15:0]=0x4000 in bits[63:48]
    g1[2] = 0x00010000;  // tensor_dim0[31:16]=0 ; tensor_dim1[15:0]=1
    g1[3] = 0x40000000;  // tensor_dim1[31:16]=0 ; tile_dim0=0x4000
    g1[4] = 0x00000001;  // tile_dim1=1 ; tile_dim2=0
    g1[5] = 16384;       // tensor_dim0_stride[31:0]
    g1[6] = 0;           // stride0 hi / stride1 lo
    g1[7] = 0;
    v4i_t gz = {0, 0, 0, 0};
#if defined(__clang_major__) && __clang_major__ >= 23
    v8i_t gz8 = {0, 0, 0, 0, 0, 0, 0, 0};
    __builtin_amdgcn_tensor_load_to_lds(g0, g1, gz, gz, gz8, 0);
#else
    __builtin_amdgcn_tensor_load_to_lds(g0, g1, gz, gz, 0);
#endif
}
#else
#define HAVE_TDM 0
#endif

// ---------------- kernel 1: per-block partial sums of query points ----------------
__global__ void k_mean_partial(const float* __restrict__ q, float* __restrict__ part, int n) {
    __shared__ float sh[3 * 256];
    float sx = 0.f, sy = 0.f, sz = 0.f;
    for (int i = blockIdx.x * blockDim.x + threadIdx.x; i < n; i += gridDim.x * blockDim.x) {
        sx += q[i * 3 + 0];
        sy += q[i * 3 + 1];
        sz += q[i * 3 + 2];
    }
    sh[threadIdx.x]       = sx;
    sh[256 + threadIdx.x] = sy;
    sh[512 + threadIdx.x] = sz;
    __syncthreads();
    for (int off = 128; off > 0; off >>= 1) {
        if ((int)threadIdx.x < off) {
            sh[threadIdx.x]       += sh[threadIdx.x + off];
            sh[256 + threadIdx.x] += sh[256 + threadIdx.x + off];
            sh[512 + threadIdx.x] += sh[512 + threadIdx.x + off];
        }
        __syncthreads();
    }
    if (threadIdx.x == 0) {
        part[blockIdx.x * 4 + 0] = sh[0];
        part[blockIdx.x * 4 + 1] = sh[256];
        part[blockIdx.x * 4 + 2] = sh[512];
    }
}

// ---------------- kernel 2: finalize mean + inverse-scale constants ----------------
__global__ void k_mean_final(const float* __restrict__ part, const float* __restrict__ scale,
                             float* __restrict__ cst, int nblocks, float invN) {
    __shared__ float sh[3 * 256];
    float sx = 0.f, sy = 0.f, sz = 0.f;
    for (int i = threadIdx.x; i < nblocks; i += 256) {
        sx += part[i * 4 + 0];
        sy += part[i * 4 + 1];
        sz += part[i * 4 + 2];
    }
    sh[threadIdx.x]       = sx;
    sh[256 + threadIdx.x] = sy;
    sh[512 + threadIdx.x] = sz;
    __syncthreads();
    for (int off = 128; off > 0; off >>= 1) {
        if ((int)threadIdx.x < off) {
            sh[threadIdx.x]       += sh[threadIdx.x + off];
            sh[256 + threadIdx.x] += sh[256 + threadIdx.x + off];
            sh[512 + threadIdx.x] += sh[512 + threadIdx.x + off];
        }
        __syncthreads();
    }
    if (threadIdx.x == 0) {
        cst[0] = sh[0]   * invN;          // mean x
        cst[1] = sh[256] * invN;          // mean y
        cst[2] = sh[512] * invN;          // mean z
        cst[3] = 2.0f / scale[0];         // 1/(scale*0.5)
        cst[4] = 2.0f / scale[1];
        cst[5] = 2.0f / scale[2];
    }
}

// ---------------- kernel 3: per-joint tri-plane bilinear sampling ----------------
// grid = (tiles, J); block = 256 (8 wave32); 192 KB dynamic LDS = 3 planes.
__global__ void __launch_bounds__(256, 1)
k_sample(const int* __restrict__ idp,
         const float* __restrict__ q,
         const float* __restrict__ cst,
         const float* __restrict__ px,
         const float* __restrict__ py,
         const float* __restrict__ pz,
         float* __restrict__ out, int npts) {
    extern __shared__ __align__(16) float lds_raw[];

    const int j  = blockIdx.y;
    const int id = idp[0];
    const size_t plane_off = ((size_t)(j * PPV + id)) * PLANE_F;

    // ---- stage 3 x 64KB planes into LDS via Tensor Data Mover ----
#if HAVE_TDM
    if (threadIdx.x == 0) {
        tdm_copy_plane(px + plane_off, 0u);
        tdm_copy_plane(py + plane_off, (unsigned)(PLANE_F * 4));
        tdm_copy_plane(pz + plane_off, (unsigned)(2 * PLANE_F * 4));
        __builtin_amdgcn_s_wait_tensorcnt(0);
    }
#else
    {
        v4f* dst = (v4f*)lds_raw;
        const v4f* s0 = (const v4f*)(px + plane_off);
        const v4f* s1 = (const v4f*)(py + plane_off);
        const v4f* s2 = (const v4f*)(pz + plane_off);
        for (int i = threadIdx.x; i < PLANE_F / 4; i += 256) {
            dst[i]                    = s0[i];
            dst[PLANE_F / 4 + i]      = s1[i];
            dst[2 * (PLANE_F / 4) + i] = s2[i];
        }
    }
#endif
    __syncthreads();

    const float mx = cst[0], my = cst[1], mz = cst[2];
    const float ivx = cst[3], ivy = cst[4], ivz = cst[5];

    const int cg   = threadIdx.x & 3;   // channel group (4 floats each)
    const int slot = threadIdx.x >> 2;  // point slot 0..63 within block
    const int tile_base = blockIdx.x * PTS_PER_BLOCK;
    const v4f* ldsv = (const v4f*)lds_raw;

    for (int it = 0; it < PTS_PER_BLOCK / 64; ++it) {
        int n  = tile_base + it * 64 + slot;
        int nc = n < npts ? n : npts - 1;

        const float X = (q[nc * 3 + 0] - mx) * ivx;
        const float Y = (q[nc * 3 + 1] - my) * ivy;
        const float Z = (q[nc * 3 + 2] - mz) * ivz;

        // plane p: grids[:, gy, gx, :]; (gy,gx) = (X,Y), (Y,Z), (Z,X)
        const float gys[3] = {X, Y, Z};
        const float gxs[3] = {Y, Z, X};

        const size_t obase = (size_t)n * OUTC + (size_t)j * (3 * CC) + (size_t)(cg * 4);

#pragma unroll
        for (int p = 0; p < 3; ++p) {
            float ix = (gxs[p] + 1.0f) * 0.5f * (float)(LL - 1);
            float iy = (gys[p] + 1.0f) * 0.5f * (float)(LL - 1);
            ix = fminf(fmaxf(ix, 0.0f), (float)(LL - 1));
            iy = fminf(fmaxf(iy, 0.0f), (float)(LL - 1));
            const float fx0 = floorf(ix), fy0 = floorf(iy);
            const int x0 = (int)fx0, y0 = (int)fy0;
            int x1 = x0 + 1; if (x1 > LL - 1) x1 = LL - 1;
            int y1 = y0 + 1; if (y1 > LL - 1) y1 = LL - 1;
            const float wx = ix - fx0, wy = iy - fy0;

            const int b = p * (PLANE_F / 4) + cg;   // v4f index base for this plane/chgroup
            const v4f v00 = ldsv[b + (((y0 << 5) + x0) << 2)];
            const v4f v01 = ldsv[b + (((y0 << 5) + x1) << 2)];
            const v4f v10 = ldsv[b + (((y1 << 5) + x0) << 2)];
            const v4f v11 = ldsv[b + (((y1 << 5) + x1) << 2)];

            const v4f top = v00 + (v01 - v00) * wx;
            const v4f bot = v10 + (v11 - v10) * wx;
            const v4f val = top + (bot - top) * wy;

            if (n < npts) {
                __builtin_nontemporal_store(val, (v4f*)(out + obase + (size_t)(p * CC)));
            }
        }
    }
}

// ---------------- host launch ----------------
extern "C" void kernel_launch(void* const* d_in, const int* in_sizes, int n_in,
                              void* d_out, int out_size, void* d_ws, size_t ws_size,
                              hipStream_t stream) {
    const int*   idp   = (const int*)d_in[0];
    const float* q     = (const float*)d_in[1];
    const float* scale = (const float*)d_in[2];
    const float* px    = (const float*)d_in[3];
    const float* py    = (const float*)d_in[4];
    const float* pz    = (const float*)d_in[5];
    float*       out   = (float*)d_out;

    const int npts = in_sizes[1] / 3;

    float* cst  = (float*)d_ws;        // 6 floats: mean[3], 2/scale[3]
    float* part = (float*)d_ws + 16;   // RED_BLOCKS * 4 floats of partial sums

    k_mean_partial<<<RED_BLOCKS, 256, 0, stream>>>(q, part, npts);
    k_mean_final<<<1, 256, 0, stream>>>(part, scale, cst, RED_BLOCKS, 1.0f / (float)npts);

    dim3 grid((npts + PTS_PER_BLOCK - 1) / PTS_PER_BLOCK, JJ);
    k_sample<<<grid, 256, 3 * PLANE_F * sizeof(float), stream>>>(idp, q, cst, px, py, pz, out, npts);
}